// DynamicGate_84043920048255
// MI455X (gfx1250) — compile-verified
//
#include <hip/hip_runtime.h>

#define C_DIM 256
#define HW_DIM 4096
#define B_DIM 32
#define SPLIT 32
#define CHUNK (HW_DIM / SPLIT)   // 128 spatial positions per partial block

typedef __attribute__((ext_vector_type(2))) float v2f;
typedef __attribute__((ext_vector_type(8))) float v8f;

// ---------------------------------------------------------------------------
// Kernel 1: partial avg-sum and max over a spatial chunk.
// grid = B*SPLIT blocks, block = 256 threads (one per channel).
// Each iteration the block reads 256 contiguous floats (1 KB) -> coalesced.
// ---------------------------------------------------------------------------
__global__ void pool_partials_kernel(const float* __restrict__ in,
                                     float* __restrict__ psum,
                                     float* __restrict__ pmax) {
  int b = blockIdx.x / SPLIT;
  int s = blockIdx.x % SPLIT;
  int c = threadIdx.x;
  const float* base = in + ((size_t)b * HW_DIM + (size_t)s * CHUNK) * C_DIM + c;
  float sum = 0.0f;
  float mx  = -__builtin_inff();
#pragma unroll 8
  for (int p = 0; p < CHUNK; ++p) {
    float v = base[(size_t)p * C_DIM];
    sum += v;
    mx = fmaxf(mx, v);
  }
  int o = blockIdx.x * C_DIM + c;
  psum[o] = sum;
  pmax[o] = mx;
}

// ---------------------------------------------------------------------------
// Kernel 2: reduce partials, blend avg/max with softmax(factor) -> pool (B,C)
// ---------------------------------------------------------------------------
__global__ void pool_finalize_kernel(const float* __restrict__ psum,
                                     const float* __restrict__ pmax,
                                     const float* __restrict__ factor,
                                     float* __restrict__ pool) {
  int b = blockIdx.x;
  int c = threadIdx.x;
  float sum = 0.0f, mx = -__builtin_inff();
#pragma unroll 8
  for (int s = 0; s < SPLIT; ++s) {
    int o = (b * SPLIT + s) * C_DIM + c;
    sum += psum[o];
    mx = fmaxf(mx, pmax[o]);
  }
  float avg = sum * (1.0f / (float)HW_DIM);
  float x0 = factor[0], x1 = factor[1];
  float m  = fmaxf(x0, x1);
  float e0 = __expf(x0 - m), e1 = __expf(x1 - m);
  float inv = 1.0f / (e0 + e1);
  pool[b * C_DIM + c] = avg * (e0 * inv) + mx * (e1 * inv);
}

// ---------------------------------------------------------------------------
// Kernel 3: h = pool @ W1 + b1 ; BN ; ReLU ; gumbel hard gate.
// Full-precision fp32 WMMA: V_WMMA_F32_16X16X4_F32, one wave per 16x16 tile.
// grid = 32 blocks x 32 threads (1 wave, EXEC all ones as WMMA requires).
// A(16x4): lane<16 holds row M, K=k..k+1; lane>=16 holds K=k+2..k+3.
// B(4x16): VGPR v = row K (lanes 0-15) / K+2 (lanes 16-31), col N = lane%16.
// C/D(16x16) v8f: VGPR j -> rows M=j (lanes 0-15) and M=j+8 (lanes 16-31).
// ---------------------------------------------------------------------------
__global__ void gate_gemm_kernel(const float* __restrict__ pool,
                                 const float* __restrict__ W1,
                                 const float* __restrict__ b1,
                                 const float* __restrict__ gamma,
                                 const float* __restrict__ beta,
                                 const float* __restrict__ bn_mean,
                                 const float* __restrict__ bn_var,
                                 const float* __restrict__ Wg,
                                 const float* __restrict__ bg,
                                 const float* __restrict__ gumbel,
                                 float* __restrict__ gate,
                                 float* __restrict__ gate_out) {
  const int lane = threadIdx.x;      // 0..31
  const int half = lane >> 4;        // 0: K+0..1, 1: K+2..3
  const int l    = lane & 15;
  const int m0 = (blockIdx.x >> 4) << 4;   // 0 or 16
  const int n0 = (blockIdx.x & 15) << 4;   // 0..240
  const int M = m0 + l;
  const int N = n0 + l;

  v8f acc = {0.f, 0.f, 0.f, 0.f, 0.f, 0.f, 0.f, 0.f};
  const float* arow = pool + (size_t)M * C_DIM + half * 2;
#pragma unroll 4
  for (int k = 0; k < C_DIM; k += 4) {
    v2f a, bm;
    a.x  = arow[k];
    a.y  = arow[k + 1];
    bm.x = W1[(size_t)(k + half * 2 + 0) * C_DIM + N];
    bm.y = W1[(size_t)(k + half * 2 + 1) * C_DIM + N];
    acc = __builtin_amdgcn_wmma_f32_16x16x4_f32(false, a, false, bm,
                                                (short)0, acc, false, false);
  }

  // Per-output-channel parameters (constant per lane: column N).
  float bb = b1[N];
  float ga = gamma[N], be = beta[N];
  float mu = bn_mean[N];
  float iv = rsqrtf(bn_var[N] + 1.0e-3f);   // BN_EPS
  float wg0 = Wg[N * 2 + 0], wg1 = Wg[N * 2 + 1];
  float bg0 = bg[N * 2 + 0], bg1 = bg[N * 2 + 1];

#pragma unroll
  for (int j = 0; j < 8; ++j) {
    int m = m0 + j + half * 8;
    float h = acc[j] + bb;
    h = ga * (h - mu) * iv + be;
    h = fmaxf(h, 0.0f);
    float g0 = gumbel[(size_t)m * (C_DIM * 2) + N * 2 + 0];
    float g1 = gumbel[(size_t)m * (C_DIM * 2) + N * 2 + 1];
    float l0 = h * wg0 + bg0 + g0;
    float l1 = h * wg1 + bg1 + g1;
    // hard straight-through: forward value is exactly one_hot(argmax)[1];
    // argmax tie -> index 0, so strict '>' here.
    float gv = (l1 > l0) ? 1.0f : 0.0f;
    gate[m * C_DIM + N]     = gv;   // scratch copy for the apply pass
    gate_out[m * C_DIM + N] = gv;   // second output tensor (B,1,1,C)
  }
}

// ---------------------------------------------------------------------------
// Kernel 4: out = in * gate (broadcast over spatial), float4 vectorized.
// 2^23 float4 elements; 2^18 float4 per image; 64 float4 per channel row.
// ---------------------------------------------------------------------------
__global__ void apply_gate_kernel(const float4* __restrict__ in,
                                  const float4* __restrict__ gate4,
                                  float4* __restrict__ out) {
  int idx = blockIdx.x * blockDim.x + threadIdx.x;
  int b  = idx >> 18;   // / (HW*C/4)
  int c4 = idx & 63;    // channel group within row
  float4 g = gate4[b * 64 + c4];
  float4 v = in[idx];
  v.x *= g.x; v.y *= g.y; v.z *= g.z; v.w *= g.w;
  out[idx] = v;
}

extern "C" void kernel_launch(void* const* d_in, const int* in_sizes, int n_in,
                              void* d_out, int out_size, void* d_ws, size_t ws_size,
                              hipStream_t stream) {
  const float* in      = (const float*)d_in[0];
  const float* factor  = (const float*)d_in[1];
  const float* W1      = (const float*)d_in[2];
  const float* b1      = (const float*)d_in[3];
  const float* gamma   = (const float*)d_in[4];
  const float* beta    = (const float*)d_in[5];
  const float* bn_mean = (const float*)d_in[6];
  const float* bn_var  = (const float*)d_in[7];
  const float* Wg      = (const float*)d_in[8];
  const float* bg      = (const float*)d_in[9];
  const float* gumbel  = (const float*)d_in[10];

  float* out_x    = (float*)d_out;
  float* out_gate = out_x + (size_t)B_DIM * HW_DIM * C_DIM;   // (B,1,1,C) tail

  float* ws   = (float*)d_ws;
  float* psum = ws;                              // B*SPLIT*C = 262144 floats
  float* pmax = psum + B_DIM * SPLIT * C_DIM;    // 262144 floats
  float* pool = pmax + B_DIM * SPLIT * C_DIM;    // B*C = 8192 floats
  float* gate = pool + B_DIM * C_DIM;            // B*C = 8192 floats

  pool_partials_kernel<<<B_DIM * SPLIT, C_DIM, 0, stream>>>(in, psum, pmax);
  pool_finalize_kernel<<<B_DIM, C_DIM, 0, stream>>>(psum, pmax, factor, pool);
  gate_gemm_kernel<<<32, 32, 0, stream>>>(pool, W1, b1, gamma, beta, bn_mean,
                                          bn_var, Wg, bg, gumbel, gate, out_gate);
  int n4 = (B_DIM * HW_DIM * C_DIM) / 4;         // 8388608
  apply_gate_kernel<<<n4 / 256, 256, 0, stream>>>((const float4*)in,
                                                  (const float4*)gate,
                                                  (float4*)out_x);
}